// EssentialMatrixSolver_13494787244420
// MI455X (gfx1250) — compile-verified
//
#include <hip/hip_runtime.h>

// CDNA5 / gfx1250 fused essential-matrix solver.
// Per batch item: G = (w*vand)^T (w*vand) via V_WMMA_F32_16X16X4_F32,
// then 9x9 Jacobi eigensolver in LDS, then small 3x3 chains.
// One wave32 per batch item; 8 waves per block; no block-level barriers.

typedef __attribute__((ext_vector_type(2))) float v2f;
typedef __attribute__((ext_vector_type(8))) float v8f;

#define NPTS 512
#define WAVES_PER_BLOCK 8
#define A9_OFF 512            // 9x9 Jacobi working matrix (after 32x16 stage)
#define V9_OFF (512 + 96)     // 9x9 eigenvector accumulator
#define WAVE_LDS 704          // floats per wave (2816 B); 8 waves = 22.5 KB

// Intra-wave LDS producer->consumer sync: wave32 lanes are lockstep, so once
// all outstanding DS ops retire (DScnt==0), every lane sees every lane's data.
__device__ __forceinline__ void wave_sync_lds() {
  asm volatile("s_wait_dscnt 0" ::: "memory");
}

__global__ __launch_bounds__(256) void ess_solver_kernel(
    const float* __restrict__ pts,      // [B, N, 4]  (x, y, u, v)
    const float* __restrict__ wts,      // [B, N]
    const float* __restrict__ normM,    // [B, 3, 3]
    const float* __restrict__ denormM,  // [B, 3, 3]
    const float* __restrict__ kM,       // [3, 3]
    float* __restrict__ out_f,          // [B, 9]
    float* __restrict__ out_e,          // [B, 9]
    float* __restrict__ out_s,          // [B, 9]
    int B) {
  __shared__ float lds[WAVES_PER_BLOCK * WAVE_LDS];
  const int tid = threadIdx.x;
  const int wave = tid >> 5;
  const int lane = tid & 31;
  const int b = blockIdx.x * WAVES_PER_BLOCK + wave;
  if (b >= B) return;
  float* wl = &lds[wave * WAVE_LDS];

  // ---------------- Phase 1: Gram matrix via f32 WMMA ----------------
  // Stage 32 rows x 16 feats of wv in LDS, then 8 accumulating
  // V_WMMA_F32_16X16X4_F32 ops (K=4 rows each). A[m,k]=wv[k,m] and
  // B[k,n]=wv[k,n] both read wv[row, feature=lane&15], so one v2f feeds both.
  v8f acc = {};                       // 16x16 f32 accumulator (G padded)
  const int feat = lane & 15;
  const int rsel = (lane >> 4) * 2;   // K-half select per ISA f32 A layout

  for (int c = 0; c < NPTS / 32; ++c) {
    const int n = c * 32 + lane;
    const float* p = pts + ((size_t)b * NPTS + n) * 4;
    const float x = p[0], y = p[1], u = p[2], v = p[3];
    const float w = wts[(size_t)b * NPTS + n];
    float4* st = (float4*)&wl[lane * 16];
    st[0] = make_float4(w * x * u, w * x * v, w * x, w * y * u);
    st[1] = make_float4(w * y * v, w * y, w * u, w * v);
    st[2] = make_float4(w, 0.0f, 0.0f, 0.0f);   // feats 9..15 padded to 0
    st[3] = make_float4(0.0f, 0.0f, 0.0f, 0.0f);
    wave_sync_lds();
#pragma unroll
    for (int r4 = 0; r4 < 8; ++r4) {
      const int row0 = r4 * 4 + rsel;
      v2f ab;
      ab.x = wl[row0 * 16 + feat];
      ab.y = wl[(row0 + 1) * 16 + feat];
      acc = __builtin_amdgcn_wmma_f32_16x16x4_f32(
          false, ab, false, ab, (short)0, acc, false, false);
    }
    wave_sync_lds();  // stage reads done before next chunk overwrites it
  }

  // ---------------- Phase 2: extract 9x9 G, init V = I ----------------
  // C/D layout: lanes 0-15 vgpr v -> G[v][lane]; lanes 16-31 -> G[v+8][lane-16]
  {
    const int col = lane & 15;
    const int mb = (lane >> 4) * 8;
    if (col < 9) {
#pragma unroll
      for (int vv = 0; vv < 8; ++vv) {
        const int m = mb + vv;
        if (m < 9) wl[A9_OFF + m * 9 + col] = acc[vv];
      }
    }
    for (int i = lane; i < 81; i += 32)
      wl[V9_OFF + i] = ((i / 9) == (i % 9)) ? 1.0f : 0.0f;
    wave_sync_lds();
  }

  // ---------------- Phase 3: cyclic Jacobi (10 sweeps) ----------------
  float* A = &wl[A9_OFF];
  float* V = &wl[V9_OFF];
  for (int sweep = 0; sweep < 10; ++sweep) {
    for (int p = 0; p < 8; ++p) {
      for (int q = p + 1; q < 9; ++q) {
        const float apq = A[p * 9 + q];
        const float app = A[p * 9 + p];
        const float aqq = A[q * 9 + q];
        float cr, sr;
        if (fabsf(apq) > 1e-20f) {   // uniform branch (same data all lanes)
          const float tau = (aqq - app) / (2.0f * apq);
          const float t =
              (tau >= 0.0f ? 1.0f : -1.0f) /
              (fabsf(tau) + sqrtf(1.0f + tau * tau));
          cr = 1.0f / sqrtf(1.0f + t * t);
          sr = t * cr;
        } else {
          cr = 1.0f;
          sr = 0.0f;
        }
        const int j = lane;
        if (j < 9) {  // row update: A <- J^T A
          const float apj = A[p * 9 + j], aqj = A[q * 9 + j];
          A[p * 9 + j] = cr * apj - sr * aqj;
          A[q * 9 + j] = sr * apj + cr * aqj;
        }
        wave_sync_lds();
        if (j < 9) {  // col update: A <- A J ;  V <- V J
          const float ajp = A[j * 9 + p], ajq = A[j * 9 + q];
          A[j * 9 + p] = cr * ajp - sr * ajq;
          A[j * 9 + q] = sr * ajp + cr * ajq;
          const float vjp = V[j * 9 + p], vjq = V[j * 9 + q];
          V[j * 9 + p] = cr * vjp - sr * vjq;
          V[j * 9 + q] = sr * vjp + cr * vjq;
        }
        wave_sync_lds();
      }
    }
  }

  // ---------------- Phase 4: singular values + null vector ----------------
  float diag[9];
#pragma unroll
  for (int j = 0; j < 9; ++j) diag[j] = A[j * 9 + j];
  int qmin = 0;
  float dmin = diag[0];
#pragma unroll
  for (int j = 1; j < 9; ++j) {
    if (diag[j] < dmin) { dmin = diag[j]; qmin = j; }
  }
  if (lane < 9) {
    out_f[(size_t)b * 9 + lane] = V[lane * 9 + qmin];  // f_raw
    const float vj = diag[lane];
    int rank = 0;  // descending order rank (SVD convention)
#pragma unroll
    for (int k = 0; k < 9; ++k)
      rank += (diag[k] > vj) || (diag[k] == vj && k < lane);
    out_s[(size_t)b * 9 + rank] = sqrtf(fmaxf(vj, 0.0f));
  }

  // ---------------- Phase 5: e = K^T (D^T F N) K ----------------
  float* F = &wl[0];   // reuse stage region as 3x3 scratch
  float* T = &wl[16];
  if (lane < 9) F[lane] = V[lane * 9 + qmin];
  wave_sync_lds();
  const float* Dm = denormM + (size_t)b * 9;
  const float* Nm = normM + (size_t)b * 9;
  const int r = lane / 3, cc = lane % 3;
  if (lane < 9) {  // T1 = D^T F
    T[lane] = Dm[0 * 3 + r] * F[0 * 3 + cc] + Dm[1 * 3 + r] * F[1 * 3 + cc] +
              Dm[2 * 3 + r] * F[2 * 3 + cc];
  }
  wave_sync_lds();
  if (lane < 9) {  // fd = T1 N
    F[lane] = T[r * 3 + 0] * Nm[0 * 3 + cc] + T[r * 3 + 1] * Nm[1 * 3 + cc] +
              T[r * 3 + 2] * Nm[2 * 3 + cc];
  }
  wave_sync_lds();
  if (lane < 9) {  // T2 = K^T fd
    T[lane] = kM[0 * 3 + r] * F[0 * 3 + cc] + kM[1 * 3 + r] * F[1 * 3 + cc] +
              kM[2 * 3 + r] * F[2 * 3 + cc];
  }
  wave_sync_lds();
  if (lane < 9) {  // e = T2 K
    out_e[(size_t)b * 9 + lane] =
        T[r * 3 + 0] * kM[0 * 3 + cc] + T[r * 3 + 1] * kM[1 * 3 + cc] +
        T[r * 3 + 2] * kM[2 * 3 + cc];
  }
}

extern "C" void kernel_launch(void* const* d_in, const int* in_sizes, int n_in,
                              void* d_out, int out_size, void* d_ws,
                              size_t ws_size, hipStream_t stream) {
  const float* pts = (const float*)d_in[0];      // p2d2d_n   [B,N,4]
  const float* wts = (const float*)d_in[1];      // weights   [B,N]
  const float* normM = (const float*)d_in[2];    // norm_mat  [B,3,3]
  const float* denormM = (const float*)d_in[3];  // denorm    [B,3,3]
  const float* kM = (const float*)d_in[4];       // k_matrix  [3,3]
  const int B = in_sizes[1] / NPTS;

  float* out = (float*)d_out;  // concat: f_raw | e_denorm | s
  float* out_f = out;
  float* out_e = out + (size_t)B * 9;
  float* out_s = out + (size_t)2 * B * 9;

  const int blocks = (B + WAVES_PER_BLOCK - 1) / WAVES_PER_BLOCK;
  hipLaunchKernelGGL(ess_solver_kernel, dim3(blocks), dim3(256), 0, stream,
                     pts, wts, normM, denormM, kM, out_f, out_e, out_s, B);
}